// GraphConv_81810537054742
// MI455X (gfx1250) — compile-verified
//
#include <hip/hip_runtime.h>

#define NN        10000      // N nodes
#define C         128        // IN_C == OUT_C
#define NEG_SLOPE 0.2f
#define KT        80         // K-chunk: 80 * 125 == 10000 (no remainder)
#define ROWS      16         // output rows per block
#define AGS       68         // A-tile K-group stride in floats (64 + 4 pad)

typedef __attribute__((ext_vector_type(2))) float v2f;
typedef __attribute__((ext_vector_type(8))) float v8f;

// K-packed layout used for both LDS tiles and the pre-swizzled support buffer:
//   idx(k, c) = (k>>2)*GROUP + c*4 + (k&3)
// A lane's WMMA fragment (k = 2h, 2h+1 at fixed column) is then one aligned b64 load.

// ---------------- Kernel 1: support = X @ W + b, stored K-packed ----------------
__global__ void __launch_bounds__(256)
gcn_support(const float* __restrict__ x,   // [NN, C]
            const float* __restrict__ w,   // [C, C]
            const float* __restrict__ b,   // [C]
            float* __restrict__ sup_sw)    // K-packed [NN/4][C][4]
{
    int gid = blockIdx.x * 256 + threadIdx.x;   // grid covers NN*C exactly
    int n = gid >> 7;          // / C
    int c = gid & (C - 1);
    const float* xr = x + (long long)n * C;
    float acc = b[c];
#pragma unroll 8
    for (int j = 0; j < C; ++j)
        acc = fmaf(xr[j], w[j * C + c], acc);
    sup_sw[(n >> 2) * (C * 4) + c * 4 + (n & 3)] = acc;
}

// ------- Kernel 2: out = LeakyReLU( (sum_k att_k A_k) @ support ), fused --------
// Block: 256 threads = 8 waves. Block owns 16 output rows; wave w owns cols [16w,16w+16).
__global__ void __launch_bounds__(256)
gcn_aggregate(const float* __restrict__ adj,     // [3, NN, NN]
              const float* __restrict__ att,     // [3]
              const float* __restrict__ sup_sw,  // K-packed support
              float* __restrict__ out)           // [NN, C]
{
    __shared__ __align__(16) float As[(KT / 4) * AGS];   // mixed A tile, K-packed+padded
    __shared__ __align__(16) float Bs[KT * C];           // support tile, K-packed linear

    const int tid  = threadIdx.x;
    const int lane = tid & 31;
    const int wave = tid >> 5;                 // 0..7 -> 16-col tile
    const int r0   = blockIdx.x * ROWS;        // output row base
    const int c0   = wave * 16;                // output col base

    const float att0 = att[0];
    const float att1 = att[1];
    const float att2 = att[2];

    const float* A0 = adj + (long long)r0 * NN;
    const float* A1 = A0 + (long long)NN * NN;
    const float* A2 = A1 + (long long)NN * NN;

    const int m = lane & 15;                   // A row / B col within tile
    const int h = lane >> 4;                   // lane-half -> K pairing
    const int aBase = m * 4 + 2 * h;           // + kg*AGS   -> {A[m][4kg+2h], +1}
    const int bBase = (c0 + m) * 4 + 2 * h;    // + kg*C*4   -> {B[4kg+2h][c], +1}

    v8f acc = {};

    for (int kb = 0; kb < NN; kb += KT) {
        // ---- stage att-mixed adjacency tile (streamed once: non-temporal) ----
#pragma unroll
        for (int j = 0; j < (ROWS * KT) / 256; ++j) {    // 5 elems / thread
            int e   = tid + j * 256;
            int row = e / KT;
            int col = e - row * KT;
            long long g = (long long)row * NN + (kb + col);
            float v = att0 * __builtin_nontemporal_load(A0 + g)
                    + att1 * __builtin_nontemporal_load(A1 + g)
                    + att2 * __builtin_nontemporal_load(A2 + g);
            As[(col >> 2) * AGS + row * 4 + (col & 3)] = v;
        }
        // ---- stage support tile: identical K-packed layout -> straight copy ----
        const float4* sb = (const float4*)(sup_sw + (long long)kb * C);
        float4* bs4 = (float4*)Bs;
#pragma unroll
        for (int j = 0; j < (KT * C) / (256 * 4); ++j)   // 10 x float4 / thread
            bs4[tid + j * 256] = sb[tid + j * 256];
        __syncthreads();

        // ---- 20 chained fp32 WMMAs: one b64 LDS load per fragment ----
#pragma unroll
        for (int kg = 0; kg < KT / 4; ++kg) {
            v2f a  = *(const v2f*)&As[kg * AGS + aBase];
            v2f bf = *(const v2f*)&Bs[kg * (C * 4) + bBase];
            acc = __builtin_amdgcn_wmma_f32_16x16x4_f32(
                false, a, false, bf, (short)0, acc, false, false);
        }
        __syncthreads();
    }

    // ---- epilogue: LeakyReLU + store. D: VGPR r -> row (r + 8h), col (c0 + m) ----
#pragma unroll
    for (int r = 0; r < 8; ++r) {
        float v = acc[r];
        v = (v >= 0.0f) ? v : NEG_SLOPE * v;
        out[(long long)(r0 + r + 8 * h) * C + (c0 + m)] = v;
    }
}

extern "C" void kernel_launch(void* const* d_in, const int* in_sizes, int n_in,
                              void* d_out, int out_size, void* d_ws, size_t ws_size,
                              hipStream_t stream) {
    const float* x   = (const float*)d_in[0];   // [10000,128]
    const float* adj = (const float*)d_in[1];   // [3,10000,10000]
    const float* att = (const float*)d_in[2];   // [3]
    const float* w   = (const float*)d_in[3];   // [128,128]
    const float* b   = (const float*)d_in[4];   // [128]
    float* out    = (float*)d_out;              // [10000,128]
    float* sup_sw = (float*)d_ws;               // 5.12 MB K-packed support scratch

    gcn_support<<<(NN * C) / 256, 256, 0, stream>>>(x, w, b, sup_sw);
    gcn_aggregate<<<NN / ROWS, 256, 0, stream>>>(adj, att, sup_sw, out);
}